// EfficientHyperbolicTripletLoss_16887811408099
// MI455X (gfx1250) — compile-verified
//
#include <hip/hip_runtime.h>

typedef float v2f __attribute__((ext_vector_type(2)));
typedef float v8f __attribute__((ext_vector_type(8)));

#define NT_TOT 655360   // N * T = 131072 * 5
#define DIM    64
#define EPSF   1e-7f

// ---------------------------------------------------------------------------
// workspace: ws[0] = float loss-sum, ((int*)ws)[1] = active count
// ---------------------------------------------------------------------------
__global__ void ehtl_init_ws(float* ws) {
    ws[0] = 0.0f;
    ((int*)ws)[1] = 0;
}

// ---------------------------------------------------------------------------
// Main kernel: 256 threads = 8 waves; each wave handles 16 triplets.
// Lane (m, h) with m = lane&15, h = lane>>4 owns row m, K-halves {4j+2h, 4j+2h+1}.
// This matches the V_WMMA_F32_16X16X4_F32 A-operand layout:
//   lanes 0-15 : VGPR0=K0, VGPR1=K1 ; lanes 16-31 : VGPR0=K2, VGPR1=K3.
// B = all-ones (layout-invariant), so D columns are the K-rowsums:
//   D[r][*] = sum_k A[r][k]  ->  squared distances, exact fp32 accumulation.
// ---------------------------------------------------------------------------
__global__ __launch_bounds__(256) void ehtl_triplet_kernel(
    const float* __restrict__ emb,
    const int*   __restrict__ aidx,
    const int*   __restrict__ pidx,
    const int*   __restrict__ nidx,
    float*       __restrict__ ws)
{
    __shared__ float red[8][32];   // [wave][0..15]=pd2, [16..31]=nd2
    __shared__ float wsum[8];
    __shared__ int   wcnt[8];

    const int lane = threadIdx.x & 31;
    const int wave = threadIdx.x >> 5;
    const int m    = lane & 15;
    const int h    = lane >> 4;            // 0: K%4 in {0,1}, 1: K%4 in {2,3}

    const int t = blockIdx.x * 128 + wave * 16 + m;   // triplet index (grid exactly covers NT)

    const int ia = aidx[t];
    const int ip = pidx[t];
    const int iq = nidx[t];

    const float* ra = emb + (size_t)ia * DIM + 2 * h;
    const float* rp = emb + (size_t)ip * DIM + 2 * h;
    const float* rn = emb + (size_t)iq * DIM + 2 * h;

    v8f cpos = {};
    v8f cneg = {};
    const v2f ones = {1.0f, 1.0f};

    float a2 = 0.0f, p2 = 0.0f, n2 = 0.0f;

#pragma unroll
    for (int j = 0; j < 16; ++j) {
        v2f av = *(const v2f*)(ra + 4 * j);
        v2f pv = *(const v2f*)(rp + 4 * j);
        v2f nv = *(const v2f*)(rn + 4 * j);

        a2 += av.x * av.x + av.y * av.y;
        p2 += pv.x * pv.x + pv.y * pv.y;
        n2 += nv.x * nv.x + nv.y * nv.y;

        v2f up = av - pv;
        v2f un = av - nv;
        v2f wp = up * up;      // A-operand slice for ||a-p||^2
        v2f wn = un * un;      // A-operand slice for ||a-n||^2

        cpos = __builtin_amdgcn_wmma_f32_16x16x4_f32(
                   false, wp, false, ones, (short)0, cpos, false, false);
        cneg = __builtin_amdgcn_wmma_f32_16x16x4_f32(
                   false, wn, false, ones, (short)0, cneg, false, false);
    }

    // each lane holds half of its row's norm contributions -> combine halves
    a2 += __shfl_xor(a2, 16, 32);
    p2 += __shfl_xor(p2, 16, 32);
    n2 += __shfl_xor(n2, 16, 32);

    // extract rowsums from column 0: lane 0 holds D[0..7][0], lane 16 holds D[8..15][0]
    if (lane == 0) {
#pragma unroll
        for (int r = 0; r < 8; ++r) {
            red[wave][r]      = cpos[r];
            red[wave][16 + r] = cneg[r];
        }
    }
    if (lane == 16) {
#pragma unroll
        for (int r = 0; r < 8; ++r) {
            red[wave][8 + r]  = cpos[r];
            red[wave][24 + r] = cneg[r];
        }
    }
    __syncthreads();

    float l  = 0.0f;
    int   ac = 0;
    if (h == 0) {   // lanes 0..15 finish the 16 triplets of this wave
        float pd2 = red[wave][m];
        float nd2 = red[wave][16 + m];

        float denp = fmaxf((1.0f - a2) * (1.0f - p2), EPSF);
        float denn = fmaxf((1.0f - a2) * (1.0f - n2), EPSF);
        float ep   = fmaxf(2.0f * pd2 / denp, EPSF);   // arg = 1 + e, e >= EPS
        float en   = fmaxf(2.0f * nd2 / denn, EPSF);
        // arccosh(1+e) = log1p(e + sqrt(e*(e+2)))  (cancellation-safe near 1)
        float posd = log1pf(ep + sqrtf(ep * (ep + 2.0f)));
        float negd = log1pf(en + sqrtf(en * (en + 2.0f)));

        float margin = 1.0f + 2.0f * sqrtf(a2);        // MARGIN*(1 + BF*||a||)
        l  = fmaxf(posd - negd + margin, 0.0f);
        ac = (l > 0.0f) ? 1 : 0;
    }

    // wave reduction (lanes >=16 contribute zeros)
    float s = l;
    int   c = ac;
#pragma unroll
    for (int off = 16; off >= 1; off >>= 1) {
        s += __shfl_xor(s, off, 32);
        c += __shfl_xor(c, off, 32);
    }
    if (lane == 0) { wsum[wave] = s; wcnt[wave] = c; }
    __syncthreads();

    if (threadIdx.x == 0) {
        float bs = 0.0f;
        int   bc = 0;
#pragma unroll
        for (int w = 0; w < 8; ++w) { bs += wsum[w]; bc += wcnt[w]; }
        atomicAdd(ws, bs);
        atomicAdd(((int*)ws) + 1, bc);
    }
}

// ---------------------------------------------------------------------------
// Finalize: outputs (loss, num_active, total, active_ratio, mean_pos_dist)
// ---------------------------------------------------------------------------
__global__ void ehtl_finalize(const float* ws, float* out) {
    float sum = ws[0];
    int   cnt = ((const int*)ws)[1];
    float loss = sum / (float)NT_TOT;
    out[0] = loss;
    out[1] = (float)cnt;
    out[2] = (float)NT_TOT;
    out[3] = (float)cnt / (float)NT_TOT;
    out[4] = loss;   // reference reuses all_losses.mean()
}

extern "C" void kernel_launch(void* const* d_in, const int* in_sizes, int n_in,
                              void* d_out, int out_size, void* d_ws, size_t ws_size,
                              hipStream_t stream) {
    const float* emb  = (const float*)d_in[0];
    // d_in[1] = labels (unused by the loss)
    const int*   aidx = (const int*)d_in[2];
    const int*   pidx = (const int*)d_in[3];
    const int*   nidx = (const int*)d_in[4];
    float* ws  = (float*)d_ws;
    float* out = (float*)d_out;

    ehtl_init_ws<<<1, 1, 0, stream>>>(ws);
    // 128 triplets per block (8 waves x 16), NT_TOT/128 = 5120 blocks, exact cover
    ehtl_triplet_kernel<<<NT_TOT / 128, 256, 0, stream>>>(emb, aidx, pidx, nidx, ws);
    ehtl_finalize<<<1, 1, 0, stream>>>(ws, out);
}